// VBrNerfLayer_36696200577472
// MI455X (gfx1250) — compile-verified
//
#include <hip/hip_runtime.h>
#include <stdint.h>

typedef __attribute__((ext_vector_type(2))) float v2f;
typedef __attribute__((ext_vector_type(8))) float v8f;

#define RES 128
#define NUM_FEAT 64
#define DATA_DIM 32
#define WAVES_PER_BLOCK 8
#define PTS_PER_WAVE 16

__device__ __forceinline__ uint32_t expand_bits(uint32_t v) {
  v = (v * 65537u) & 4278190335u;
  v = (v * 257u)   & 251719695u;
  v = (v * 17u)    & 3272356035u;
  v = (v * 5u)     & 1227133513u;
  return v;
}

__launch_bounds__(256)
__global__ void vqnerf_trilerp_kernel(const float* __restrict__ ray_p,
                                      const float* __restrict__ grid,
                                      const float* __restrict__ codebook,
                                      float* __restrict__ out,
                                      int n) {
  // Per-wave 16x64 fp32 coefficient tile + block-shared codebook stage.
  __shared__ float s_coeff[WAVES_PER_BLOCK][PTS_PER_WAVE * NUM_FEAT]; // 32 KB
  __shared__ float s_cb[NUM_FEAT * DATA_DIM];                         // 8 KB

  const int lane = threadIdx.x & 31;
  const int wave = threadIdx.x >> 5;

  // Stage the 8 KB codebook once per block.
  for (int i = threadIdx.x; i < NUM_FEAT * DATA_DIM; i += 256)
    s_cb[i] = codebook[i];

  // Zero this wave's coeff tile.
  float* coeff = s_coeff[wave];
#pragma unroll
  for (int i = 0; i < (PTS_PER_WAVE * NUM_FEAT) / 32; ++i)
    coeff[i * 32 + lane] = 0.0f;

  __syncthreads();

  const int base = (blockIdx.x * WAVES_PER_BLOCK + wave) * PTS_PER_WAVE;

  // ---- Phase 1: per-(point,corner) gather + softmax-max + scatter coeff ----
  // 16 points * 8 corners = 128 tasks, 4 per lane.
#pragma unroll
  for (int it = 0; it < 4; ++it) {
    const int t = it * 32 + lane;
    const int p = t >> 3;   // local point 0..15
    const int k = t & 7;    // corner 0..7
    const int pt = base + p;
    if (pt < n) {
      const float px = ray_p[pt * 3 + 0];
      const float py = ray_p[pt * 3 + 1];
      const float pz = ray_p[pt * 3 + 2];
      const float fx = floorf(px), fy = floorf(py), fz = floorf(pz);
      const float ox = px - fx, oy = py - fy, oz = pz - fz;
      const int bx = (k >> 2) & 1, by = (k >> 1) & 1, bz = k & 1;
      int cx = (int)fx + bx; cx = cx < 0 ? 0 : (cx > RES - 1 ? RES - 1 : cx);
      int cy = (int)fy + by; cy = cy < 0 ? 0 : (cy > RES - 1 ? RES - 1 : cy);
      int cz = (int)fz + bz; cz = cz < 0 ? 0 : (cz > RES - 1 ? RES - 1 : cz);
      const uint32_t mort = expand_bits((uint32_t)cx) |
                            (expand_bits((uint32_t)cy) << 1) |
                            (expand_bits((uint32_t)cz) << 2);

      // Gather one 256B grid row (single pass, kept in registers).
      const float4* row = (const float4*)(grid + (size_t)mort * NUM_FEAT);
      float v[NUM_FEAT];
#pragma unroll
      for (int j = 0; j < NUM_FEAT / 4; ++j) {
        const float4 q = row[j];
        v[4 * j + 0] = q.x; v[4 * j + 1] = q.y;
        v[4 * j + 2] = q.z; v[4 * j + 3] = q.w;
      }
      // argmax (first-max, matching jnp.argmax) and sum of exp(l - max).
      float mx = v[0]; int am = 0;
#pragma unroll
      for (int f = 1; f < NUM_FEAT; ++f)
        if (v[f] > mx) { mx = v[f]; am = f; }
      float se = 0.0f;
#pragma unroll
      for (int f = 0; f < NUM_FEAT; ++f) se += __expf(v[f] - mx);

      const float smax  = 1.0f / se;                 // softmax value at argmax
      const float scale = (1.0f - smax) + smax;      // straight-through residual
      const float wx = bx ? ox : 1.0f - ox;
      const float wy = by ? oy : 1.0f - oy;
      const float wz = bz ? oz : 1.0f - oz;

      atomicAdd(&coeff[p * NUM_FEAT + am], wx * wy * wz * scale); // ds_add_f32
    }
  }

  __syncthreads();

  // ---- Phase 2: out(16x32) = coeff(16x64) @ codebook(64x32) via chained ----
  // ---- V_WMMA_F32_16X16X4_F32 (fp32-exact, K=64 in 16 steps, 2 col tiles) --
  const int m16  = lane & 15;
  const int hi   = lane >> 4;    // 0: K+{0,1}, rows M; 1: K+{2,3}
  const int koff = hi * 2;

  // A fragments: 16x4 fp32 layout -> lane m16 holds (K=koff, K=koff+1) per step.
  v2f afrag[16];
#pragma unroll
  for (int t0 = 0; t0 < 16; ++t0) {
    const int kb = 4 * t0 + koff;
    v2f a;
    a.x = coeff[m16 * NUM_FEAT + kb];
    a.y = coeff[m16 * NUM_FEAT + kb + 1];
    afrag[t0] = a;
  }

#pragma unroll
  for (int c = 0; c < 2; ++c) {
    v8f acc = {0.0f, 0.0f, 0.0f, 0.0f, 0.0f, 0.0f, 0.0f, 0.0f};
#pragma unroll
    for (int t0 = 0; t0 < 16; ++t0) {
      const int kb = 4 * t0 + koff;
      v2f b;   // B 4x16 fp32: row K striped across lanes within a VGPR
      b.x = s_cb[kb * DATA_DIM + c * 16 + m16];
      b.y = s_cb[(kb + 1) * DATA_DIM + c * 16 + m16];
      acc = __builtin_amdgcn_wmma_f32_16x16x4_f32(
          /*neg_a=*/false, afrag[t0], /*neg_b=*/false, b,
          /*c_mod=*/(short)0, acc, /*reuse_a=*/false, /*reuse_b=*/false);
    }
    // D layout: VGPR r -> row M = r + 8*hi, col N = lane&15 (+ col tile).
    const int col = c * 16 + m16;
    const int mb  = hi * 8;
#pragma unroll
    for (int r = 0; r < 8; ++r) {
      const int pt = base + mb + r;
      if (pt < n) out[pt * DATA_DIM + col] = acc[r];
    }
  }
}

extern "C" void kernel_launch(void* const* d_in, const int* in_sizes, int n_in,
                              void* d_out, int out_size, void* d_ws, size_t ws_size,
                              hipStream_t stream) {
  const float* ray_p    = (const float*)d_in[0];
  const float* grid     = (const float*)d_in[1];
  const float* codebook = (const float*)d_in[2];
  float* out = (float*)d_out;

  const int n = in_sizes[0] / 3;
  const int tiles  = (n + PTS_PER_WAVE - 1) / PTS_PER_WAVE;
  const int blocks = (tiles + WAVES_PER_BLOCK - 1) / WAVES_PER_BLOCK;

  hipLaunchKernelGGL(vqnerf_trilerp_kernel, dim3(blocks), dim3(256), 0, stream,
                     ray_p, grid, codebook, out, n);
}